// GCN_54906861912525
// MI455X (gfx1250) — compile-verified
//
#include <hip/hip_runtime.h>

typedef __attribute__((ext_vector_type(2))) float v2f;
typedef __attribute__((ext_vector_type(8))) float v8f;

#define DIM 64

// ---------------------------------------------------------------- zero fill
__global__ void zero_f32(float* __restrict__ p, long long n) {
  long long i = (long long)blockIdx.x * blockDim.x + threadIdx.x;
  if (i < n) p[i] = 0.0f;
}

// ------------------------------------------------- edge scatter (mean-agg)
// 64 threads per edge (one wave = 32 contiguous dims of one edge ->
// coalesced 128B read from x, contiguous 128B atomic region on agg).
// agg fits in L2 (25.6MB) so atomics resolve at L2.
__global__ void sage_scatter(const float* __restrict__ x,
                             const long long* __restrict__ ei,
                             float* __restrict__ agg,
                             float* __restrict__ deg,
                             long long E) {
  long long tid = (long long)blockIdx.x * blockDim.x + threadIdx.x;
  long long e = tid >> 6;
  if (e >= E) return;
  int d = (int)(tid & 63);
  long long src = ei[e];
  long long dst = ei[E + e];
  float v = x[src * DIM + d];
  atomicAdd(&agg[dst * DIM + d], v);
  if (d == 0) atomicAdd(&deg[dst], 1.0f);
}

// ------------------------------------- fused mean / GEMMx2 / bias / lsm
// One wave handles 16 nodes. D-tile: 16 nodes x 64 cols = four 16x16 f32
// accumulators. K=128 total (mean|x concat) as 32 V_WMMA_F32_16X16X4_F32
// steps, 4 independent chains per step.
//
// f32 16x4 A layout: lanes 0-15 hold M=0..15 with K=k,k+1; lanes 16-31 the
// same rows with K=k+2,k+3 -> per-lane float2 at (row, k + (lane>>4)*2).
// B = W^T (B[k][n] = W[n][k]) mirrors that layout with N = lane&15.
// C/D layout: VGPR j holds row M=j (lanes 0-15) / M=j+8 (lanes 16-31),
// N = lane&15 -> row reductions use shfl_xor masks 1,2,4,8 (stay in half).
__global__ void __launch_bounds__(256)
sage_wmma(const float* __restrict__ x,
          const float* __restrict__ agg,
          const float* __restrict__ deg,
          const float* __restrict__ Wl,   // [64][64] row-major
          const float* __restrict__ bl,   // [64]
          const float* __restrict__ Wr,   // [64][64] row-major
          float* __restrict__ out,
          float* __restrict__ z,
          int nTiles, int N) {
  int lane = threadIdx.x & 31;
  int wave = threadIdx.x >> 5;
  int tile = blockIdx.x * 8 + wave;
  if (tile >= nTiles) return;           // wave-uniform: EXEC stays all-1s

  int n15  = lane & 15;
  int half = lane >> 4;                 // 0 or 1
  int koff = half * 2;

  int rowIdx = tile * 16 + n15;
  int rowC   = rowIdx < N ? rowIdx : N - 1;  // clamp reads only

  float dg   = deg[rowC];
  float rdeg = 1.0f / fmaxf(dg, 1.0f);

  const float* aRow = agg + (long long)rowC * DIM;
  const float* xRow = x   + (long long)rowC * DIM;

  v8f acc0 = {}, acc1 = {}, acc2 = {}, acc3 = {};

  // ---- mean @ Wl^T ----
  for (int kk = 0; kk < 16; ++kk) {
    int k = kk * 4 + koff;
    v2f a = *(const v2f*)(aRow + k);
    a = a * rdeg;
    v2f b0 = *(const v2f*)(Wl + ( 0 + n15) * DIM + k);
    v2f b1 = *(const v2f*)(Wl + (16 + n15) * DIM + k);
    v2f b2 = *(const v2f*)(Wl + (32 + n15) * DIM + k);
    v2f b3 = *(const v2f*)(Wl + (48 + n15) * DIM + k);
    acc0 = __builtin_amdgcn_wmma_f32_16x16x4_f32(false, a, false, b0, (short)0, acc0, false, false);
    acc1 = __builtin_amdgcn_wmma_f32_16x16x4_f32(false, a, false, b1, (short)0, acc1, false, false);
    acc2 = __builtin_amdgcn_wmma_f32_16x16x4_f32(false, a, false, b2, (short)0, acc2, false, false);
    acc3 = __builtin_amdgcn_wmma_f32_16x16x4_f32(false, a, false, b3, (short)0, acc3, false, false);
  }
  // ---- x @ Wr^T ----
  for (int kk = 0; kk < 16; ++kk) {
    int k = kk * 4 + koff;
    v2f a = *(const v2f*)(xRow + k);
    v2f b0 = *(const v2f*)(Wr + ( 0 + n15) * DIM + k);
    v2f b1 = *(const v2f*)(Wr + (16 + n15) * DIM + k);
    v2f b2 = *(const v2f*)(Wr + (32 + n15) * DIM + k);
    v2f b3 = *(const v2f*)(Wr + (48 + n15) * DIM + k);
    acc0 = __builtin_amdgcn_wmma_f32_16x16x4_f32(false, a, false, b0, (short)0, acc0, false, false);
    acc1 = __builtin_amdgcn_wmma_f32_16x16x4_f32(false, a, false, b1, (short)0, acc1, false, false);
    acc2 = __builtin_amdgcn_wmma_f32_16x16x4_f32(false, a, false, b2, (short)0, acc2, false, false);
    acc3 = __builtin_amdgcn_wmma_f32_16x16x4_f32(false, a, false, b3, (short)0, acc3, false, false);
  }

  // ---- bias (constant per lane/column) ----
  float bias0 = bl[ 0 + n15], bias1 = bl[16 + n15];
  float bias2 = bl[32 + n15], bias3 = bl[48 + n15];
#pragma unroll
  for (int j = 0; j < 8; ++j) {
    acc0[j] += bias0; acc1[j] += bias1; acc2[j] += bias2; acc3[j] += bias3;
  }

  // ---- log_softmax over each row + store ----
  long long base = (long long)tile * 16;
#pragma unroll
  for (int j = 0; j < 8; ++j) {
    float e0 = acc0[j], e1 = acc1[j], e2 = acc2[j], e3 = acc3[j];
    float mx = fmaxf(fmaxf(e0, e1), fmaxf(e2, e3));
    mx = fmaxf(mx, __shfl_xor(mx, 1, 32));
    mx = fmaxf(mx, __shfl_xor(mx, 2, 32));
    mx = fmaxf(mx, __shfl_xor(mx, 4, 32));
    mx = fmaxf(mx, __shfl_xor(mx, 8, 32));
    float s = expf(e0 - mx) + expf(e1 - mx) + expf(e2 - mx) + expf(e3 - mx);
    s += __shfl_xor(s, 1, 32);
    s += __shfl_xor(s, 2, 32);
    s += __shfl_xor(s, 4, 32);
    s += __shfl_xor(s, 8, 32);
    float lse = mx + logf(s);

    long long orow = base + j + (half << 3);
    if (orow < N) {
      long long o = orow * DIM + n15;
      out[o +  0] = e0;        out[o + 16] = e1;
      out[o + 32] = e2;        out[o + 48] = e3;
      z[o +  0]   = e0 - lse;  z[o + 16]   = e1 - lse;
      z[o + 32]   = e2 - lse;  z[o + 48]   = e3 - lse;
    }
  }
}

// ---------------------------------------------------------------- launcher
extern "C" void kernel_launch(void* const* d_in, const int* in_sizes, int n_in,
                              void* d_out, int out_size, void* d_ws, size_t ws_size,
                              hipStream_t stream) {
  const float*     x  = (const float*)d_in[0];
  const long long* ei = (const long long*)d_in[1];  // int64 edge_index [2,E]
  const float*     Wl = (const float*)d_in[2];
  const float*     bl = (const float*)d_in[3];
  const float*     Wr = (const float*)d_in[4];

  int       N = in_sizes[0] / DIM;
  long long E = (long long)in_sizes[1] / 2;

  float* out = (float*)d_out;
  float* z   = out + (long long)N * DIM;

  size_t aggBytes = (size_t)N * DIM * sizeof(float);
  size_t degBytes = (size_t)N * sizeof(float);
  float *agg, *deg;
  if (ws_size >= aggBytes + degBytes) {
    agg = (float*)d_ws;
    deg = agg + (size_t)N * DIM;
  } else {
    // reuse z output region as accumulator; each wave reads its agg rows
    // fully before overwriting them with z.
    agg = z;
    deg = (float*)d_ws;
  }

  long long aggN = (long long)N * DIM;
  zero_f32<<<(unsigned)((aggN + 255) / 256), 256, 0, stream>>>(agg, aggN);
  zero_f32<<<(unsigned)((N + 255) / 256), 256, 0, stream>>>(deg, (long long)N);

  long long st = E * 64;
  sage_scatter<<<(unsigned)((st + 255) / 256), 256, 0, stream>>>(x, ei, agg, deg, E);

  int nTiles = (N + 15) / 16;
  int blocks = (nTiles + 7) / 8;
  sage_wmma<<<blocks, 256, 0, stream>>>(x, agg, deg, Wl, bl, Wr, out, z, nTiles, N);
}